// MyModel_21912923144628
// MI455X (gfx1250) — compile-verified
//
#include <hip/hip_runtime.h>
#include <cmath>

// ---- problem constants (from reference) ----
#define NN      50      // N == NITEM
#define DIMV    64
#define HHEAD   4
#define KASSIGN 4
#define NNEG    4
#define NC      5       // candidates = 1 pos + 4 neg
#define MAXLEN  512
#define PITCH   65      // LDS row pitch for 64-wide matrices

typedef float v2f __attribute__((ext_vector_type(2)));
typedef float v8f __attribute__((ext_vector_type(8)));

__device__ __forceinline__ v8f wmma4(v2f a, v2f b, v8f c) {
  // V_WMMA_F32_16X16X4_F32 : D = A(16x4) * B(4x16) + C(16x16)
  return __builtin_amdgcn_wmma_f32_16x16x4_f32(false, a, false, b, (short)0, c, false, false);
}

// D(64x64 LDS,pitch) = A(64x64 LDS,pitch) * B(64x64 global row-major, ld=64)
__device__ __forceinline__ void gemm_AB_g(const float* A, const float* Bg, float* D,
                                          int wave, int lane) {
  const int l15 = lane & 15;
  const int ko  = (lane >> 4) << 1;           // 0 or 2
  const int r   = wave * 16 + l15;            // A row
#pragma unroll
  for (int nt = 0; nt < 4; ++nt) {
    const int col = nt * 16 + l15;
    v8f c = {0.f, 0.f, 0.f, 0.f, 0.f, 0.f, 0.f, 0.f};
#pragma unroll
    for (int k0 = 0; k0 < 64; k0 += 4) {
      v2f a, b;
      a.x = A[r * PITCH + k0 + ko];
      a.y = A[r * PITCH + k0 + ko + 1];
      b.x = Bg[(k0 + ko) * 64 + col];
      b.y = Bg[(k0 + ko + 1) * 64 + col];
      c = wmma4(a, b, c);
    }
    const int rb = wave * 16 + ((lane >> 4) << 3);
#pragma unroll
    for (int v = 0; v < 8; ++v) D[(rb + v) * PITCH + col] = c[v];
  }
}

// D(64x64 LDS,pitch) = A(64x64 LDS,pitch) * B(64x64 LDS,pitch)
__device__ __forceinline__ void gemm_AB_l(const float* A, const float* Bl, float* D,
                                          int wave, int lane) {
  const int l15 = lane & 15;
  const int ko  = (lane >> 4) << 1;
  const int r   = wave * 16 + l15;
#pragma unroll
  for (int nt = 0; nt < 4; ++nt) {
    const int col = nt * 16 + l15;
    v8f c = {0.f, 0.f, 0.f, 0.f, 0.f, 0.f, 0.f, 0.f};
#pragma unroll
    for (int k0 = 0; k0 < 64; k0 += 4) {
      v2f a, b;
      a.x = A[r * PITCH + k0 + ko];
      a.y = A[r * PITCH + k0 + ko + 1];
      b.x = Bl[(k0 + ko) * PITCH + col];
      b.y = Bl[(k0 + ko + 1) * PITCH + col];
      c = wmma4(a, b, c);
    }
    const int rb = wave * 16 + ((lane >> 4) << 3);
#pragma unroll
    for (int v = 0; v < 8; ++v) D[(rb + v) * PITCH + col] = c[v];
  }
}

// acc[nt] += A(64x64 LDS,pitch) * B(64x64 global, ld=64)  (C carried in registers)
__device__ __forceinline__ void gemm_acc_g(const float* A, const float* Bg, v8f* acc,
                                           int wave, int lane) {
  const int l15 = lane & 15;
  const int ko  = (lane >> 4) << 1;
  const int r   = wave * 16 + l15;
#pragma unroll
  for (int nt = 0; nt < 4; ++nt) {
    const int col = nt * 16 + l15;
    v8f c = acc[nt];
#pragma unroll
    for (int k0 = 0; k0 < 64; k0 += 4) {
      v2f a, b;
      a.x = A[r * PITCH + k0 + ko];
      a.y = A[r * PITCH + k0 + ko + 1];
      b.x = Bg[(k0 + ko) * 64 + col];
      b.y = Bg[(k0 + ko + 1) * 64 + col];
      c = wmma4(a, b, c);
    }
    acc[nt] = c;
  }
}

__device__ __forceinline__ float softplusf(float x) {
  return (x > 0.f) ? (x + log1pf(expf(-x))) : log1pf(expf(x));
}

__device__ __forceinline__ void argsort5_desc(const float* v, int* ind) {
  bool used[NC] = {false, false, false, false, false};
  for (int r = 0; r < NC; ++r) {
    int best = -1; float bv = 0.f;
    for (int i = 0; i < NC; ++i)
      if (!used[i] && (best < 0 || v[i] > bv)) { best = i; bv = v[i]; }
    used[best] = true; ind[r] = best;
  }
}

// ssl_topk(user, cand[ind[0..1]], cand[ind[2..3]]), tao=0.5
__device__ __forceinline__ float ssl2(const float* user, const float* candb, const int* ind) {
  float un = 0.f;
  for (int d = 0; d < DIMV; ++d) un += user[d] * user[d];
  const float uinv = 1.f / sqrtf(un);
  float ps = 0.f, ns = 0.f;
  for (int k = 0; k < 4; ++k) {
    const float* p = candb + ind[k] * DIMV;
    float pn = 0.f, dot = 0.f;
    for (int d = 0; d < DIMV; ++d) { pn += p[d] * p[d]; dot += p[d] * user[d]; }
    const float sim = dot * uinv * (1.f / sqrtf(pn));
    const float e = expf(sim * 2.f);   // / tao(=0.5)
    if (k < 2) ps += e; else ns += e;
  }
  return -(logf(ps) - logf(ps + ns));
}

// ---------- peW[t] = pe[t] @ W_PE, precomputed once ----------
__global__ void k_prep_pe(const float* __restrict__ W_PE, float* __restrict__ peW) {
  __shared__ float pe[DIMV];
  const int t = blockIdx.x;     // 0..511
  const int j = threadIdx.x;    // 0..63
  if (j < 32) {
    const float div = expf(-(float)(2 * j) * (9.210340371976184f / 64.f)); // ln(1e4)/64
    const float ang = (float)t * div;
    pe[2 * j]     = sinf(ang);
    pe[2 * j + 1] = cosf(ang);
  }
  __syncthreads();
  float acc = 0.f;
  for (int d = 0; d < DIMV; ++d) acc += pe[d] * W_PE[d * DIMV + j];
  peW[t * DIMV + j] = acc;
}

// ---------- zero out, add weight-norm part of loss_reg ----------
__global__ void k_init(const float* __restrict__ Wpred, const float* __restrict__ WPE,
                       float* __restrict__ out) {
  __shared__ float red[256];
  const int tid = threadIdx.x;
  float s = 0.f;
  for (int i = tid; i < 128 * 64; i += 256) { float v = Wpred[i]; s += v * v; }
  for (int i = tid; i < 64 * 64;  i += 256) { float v = WPE[i];  s += v * v; }
  red[tid] = s;
  __syncthreads();
  for (int off = 128; off > 0; off >>= 1) {
    if (tid < off) red[tid] += red[tid + off];
    __syncthreads();
  }
  if (tid == 0) { out[0] = 0.f; out[1] = red[0]; out[2] = 0.f; out[3] = 0.f; }
}

// ---------- one workgroup per batch element ----------
__global__ void __launch_bounds__(128) k_main(
    const float* __restrict__ emb,   const float* __restrict__ Wgat,
    const float* __restrict__ a_src, const float* __restrict__ a_dst,
    const float* __restrict__ W_o,   const float* __restrict__ W_assign,
    const float* __restrict__ adj,   const float* __restrict__ Wpred,
    const int* __restrict__ history, const int* __restrict__ timestp,
    const int* __restrict__ pos_id,  const int* __restrict__ neg_id,
    const int* __restrict__ stp,     const float* __restrict__ peW,
    float* __restrict__ out, int B)
{
  __shared__ float xs[64 * PITCH];    // x, later reused for x_new_I
  __shared__ float whs[64 * PITCH];   // Wh for current head
  __shared__ float ats[64 * PITCH];   // attn for current head
  __shared__ float ohs[64 * PITCH];   // attn @ Wh for current head
  __shared__ float cand[NC * DIMV];
  __shared__ float xp[KASSIGN * DIMV];        // x_parent
  __shared__ float sS[64 * KASSIGN];          // assignment s
  __shared__ float esd[128];                  // es[0..63], ed[64..127]
  __shared__ float attF[NC * KASSIGN];
  __shared__ float attI[NC * 64];
  __shared__ float uF[NC * DIMV];
  __shared__ float uI[NC * DIMV];
  __shared__ float uo[NC * DIMV];
  __shared__ float userI[DIMV], userF[DIMV];
  __shared__ float sIv[NC], sFv[NC], scv[NC];
  __shared__ float red[2];                    // 0: reg-partial, 1: entropy-partial

  const int b    = blockIdx.x;
  const int tid  = threadIdx.x;
  const int lane = tid & 31;
  const int wave = tid >> 5;

  if (tid < 2) red[tid] = 0.f;
  __syncthreads();

  // ---- 1. gather x = (_x + peW[timestp]) * (history>0), accumulate ||_x||^2 ----
  float regp = 0.f;
  for (int idx = tid; idx < 64 * 64; idx += 128) {
    const int row = idx >> 6, col = idx & 63;
    float xv = 0.f;
    if (row < NN) {
      const int hid = history[b * NN + row];
      const int ts  = timestp[b * NN + row];
      const float ev = emb[hid * DIMV + col];
      regp += ev * ev;
      if (hid > 0) xv = ev + peW[ts * DIMV + col];
    }
    xs[row * PITCH + col] = xv;
  }
  // cand = [pos, neg] + peW[stp]
  const int stv = stp[b];
  for (int idx = tid; idx < NC * DIMV; idx += 128) {
    const int c = idx >> 6, col = idx & 63;
    const int id = (c == 0) ? pos_id[b] : neg_id[b * NNEG + (c - 1)];
    const float ev = emb[id * DIMV + col];
    regp += ev * ev;
    cand[idx] = ev + peW[stv * DIMV + col];
  }
  atomicAdd(&red[0], regp);
  __syncthreads();

  // ---- 2. s = softmax(x @ W_assign), entropy ----
  for (int idx = tid; idx < NN * KASSIGN; idx += 128) {
    const int n = idx >> 2, k = idx & 3;
    float acc = 0.f;
    for (int d = 0; d < DIMV; ++d) acc += xs[n * PITCH + d] * W_assign[d * KASSIGN + k];
    sS[n * KASSIGN + k] = acc;
  }
  __syncthreads();
  if (tid < NN) {
    float l[4];
    for (int k = 0; k < 4; ++k) l[k] = sS[tid * 4 + k];
    float m = fmaxf(fmaxf(l[0], l[1]), fmaxf(l[2], l[3]));
    float sum = 0.f;
    for (int k = 0; k < 4; ++k) { l[k] = expf(l[k] - m); sum += l[k]; }
    const float inv = 1.f / sum;
    float ent = 0.f;
    for (int k = 0; k < 4; ++k) {
      const float p = l[k] * inv;
      sS[tid * 4 + k] = p;
      ent -= p * logf(p);
    }
    atomicAdd(&red[1], ent);
  }
  __syncthreads();

  // ---- 3. x_parent = s^T x ----
  for (int idx = tid; idx < KASSIGN * DIMV; idx += 128) {
    const int k = idx >> 6, d = idx & 63;
    float acc = 0.f;
    for (int n = 0; n < NN; ++n) acc += sS[n * KASSIGN + k] * xs[n * PITCH + d];
    xp[idx] = acc;
  }
  __syncthreads();

  // ---- 4. GAT (heads sequential; W_o contribution kept in WMMA accumulators) ----
  v8f acc4[4];
#pragma unroll
  for (int i = 0; i < 4; ++i) acc4[i] = (v8f){0.f, 0.f, 0.f, 0.f, 0.f, 0.f, 0.f, 0.f};

  for (int h = 0; h < HHEAD; ++h) {
    gemm_AB_g(xs, Wgat + h * 64 * 64, whs, wave, lane);        // Wh = x @ W_gat[h]
    __syncthreads();
    {
      const int n = tid >> 1;
      float acc = 0.f;
      if (n < NN) {
        const float* av = (tid & 1) ? (a_dst + h * DIMV) : (a_src + h * DIMV);
        for (int d = 0; d < DIMV; ++d) acc += whs[n * PITCH + d] * av[d];
      }
      esd[((tid & 1) << 6) + n] = acc;
    }
    __syncthreads();
    if (tid < 64) {                                            // softmax rows of e
      const int n = tid;
      if (n < NN) {
        const float esn = esd[n];
        const float* adjr = adj + ((size_t)b * NN + n) * NN;
        float mx = -1e30f;
        for (int m = 0; m < NN; ++m) {
          float e = esn + esd[64 + m];
          e = (e >= 0.f) ? e : 0.2f * e;                       // leaky_relu 0.2
          e = (adjr[m] > 0.f) ? e : -1e9f;
          mx = fmaxf(mx, e);
        }
        float sum = 0.f;
        for (int m = 0; m < NN; ++m) {
          float e = esn + esd[64 + m];
          e = (e >= 0.f) ? e : 0.2f * e;
          e = (adjr[m] > 0.f) ? e : -1e9f;
          const float ex = expf(e - mx);
          sum += ex;
          ats[n * PITCH + m] = ex;
        }
        const float inv = 1.f / sum;
        for (int m = 0; m < NN; ++m) ats[n * PITCH + m] *= inv;
        for (int m = NN; m < 64; ++m) ats[n * PITCH + m] = 0.f;
      } else {
        for (int m = 0; m < 64; ++m) ats[n * PITCH + m] = 0.f;
      }
    }
    __syncthreads();
    gemm_AB_l(ats, whs, ohs, wave, lane);                      // outh = attn @ Wh
    __syncthreads();
    gemm_acc_g(ohs, W_o + h * 64 * 64, acc4, wave, lane);      // acc += outh @ W_o[h]
    __syncthreads();
  }

  // ---- 5. x_new = elu(acc) -> reuse xs ----
  {
    const int l15 = lane & 15;
    const int rb = wave * 16 + ((lane >> 4) << 3);
#pragma unroll
    for (int nt = 0; nt < 4; ++nt) {
      const int col = nt * 16 + l15;
#pragma unroll
      for (int v = 0; v < 8; ++v) {
        const float x = acc4[nt][v];
        xs[(rb + v) * PITCH + col] = (x > 0.f) ? x : (expf(x) - 1.f);
      }
    }
  }
  __syncthreads();

  // ---- 6. user vectors ----
  if (tid < DIMV) {
    float s = 0.f;
    for (int n = 0; n < NN; ++n) s += xs[n * PITCH + tid];
    userI[tid] = s * (1.f / NN);
    float f = 0.f;
    for (int k = 0; k < KASSIGN; ++k) f += xp[k * DIMV + tid];
    userF[tid] = f * 0.25f;
  }
  __syncthreads();

  // ---- 7. aggregates u_F, u_I ----
  for (int idx = tid; idx < NC * KASSIGN; idx += 128) {
    const int c = idx >> 2, k = idx & 3;
    float a = 0.f;
    for (int d = 0; d < DIMV; ++d) a += xp[k * DIMV + d] * cand[c * DIMV + d];
    attF[idx] = a;
  }
  for (int idx = tid; idx < NC * NN; idx += 128) {
    const int c = idx / NN, n = idx % NN;
    float a = 0.f;
    for (int d = 0; d < DIMV; ++d) a += xs[n * PITCH + d] * cand[c * DIMV + d];
    attI[c * 64 + n] = a;
  }
  __syncthreads();
  if (tid < NC) {
    float m = -1e30f;
    for (int k = 0; k < 4; ++k) m = fmaxf(m, attF[tid * 4 + k]);
    float s = 0.f, e[4];
    for (int k = 0; k < 4; ++k) { e[k] = expf(attF[tid * 4 + k] - m); s += e[k]; }
    for (int k = 0; k < 4; ++k) attF[tid * 4 + k] = e[k] / s;
  }
  if (tid >= 8 && tid < 8 + NC) {
    const int c = tid - 8;
    float m = -1e30f;
    for (int n = 0; n < NN; ++n) m = fmaxf(m, attI[c * 64 + n]);
    float s = 0.f;
    for (int n = 0; n < NN; ++n) { const float e = expf(attI[c * 64 + n] - m); attI[c * 64 + n] = e; s += e; }
    const float inv = 1.f / s;
    for (int n = 0; n < NN; ++n) attI[c * 64 + n] *= inv;
  }
  __syncthreads();
  for (int idx = tid; idx < NC * DIMV; idx += 128) {
    const int c = idx >> 6, d = idx & 63;
    float f = 0.f;
    for (int k = 0; k < KASSIGN; ++k) f += attF[c * 4 + k] * xp[k * DIMV + d];
    uF[idx] = f;
    float g = 0.f;
    for (int n = 0; n < NN; ++n) g += attI[c * 64 + n] * xs[n * PITCH + d];
    uI[idx] = g;
  }
  __syncthreads();

  // ---- 8. u = [uF,uI] @ W_pred ; scores; sI/sF ----
  for (int idx = tid; idx < NC * DIMV; idx += 128) {
    const int c = idx >> 6, d = idx & 63;
    float a = 0.f;
    for (int j = 0; j < 64; ++j) a += uF[c * DIMV + j] * Wpred[j * DIMV + d];
    for (int j = 0; j < 64; ++j) a += uI[c * DIMV + j] * Wpred[(64 + j) * DIMV + d];
    uo[idx] = a;
  }
  __syncthreads();
  if (tid < NC) {
    float s = 0.f;
    for (int d = 0; d < DIMV; ++d) s += uo[tid * DIMV + d] * cand[tid * DIMV + d];
    scv[tid] = s;
  }
  if (tid >= 8 && tid < 8 + 2 * NC) {
    const int c = (tid - 8) % NC, w = (tid - 8) / NC;
    const float* uvec = w ? userF : userI;
    float s = 0.f;
    for (int d = 0; d < DIMV; ++d) s += cand[c * DIMV + d] * uvec[d];
    if (w) sFv[c] = s; else sIv[c] = s;
  }
  __syncthreads();

  // ---- 9. losses ----
  if (tid == 0) {
    float l = 0.f;
    for (int c = 1; c < NC; ++c) l += softplusf(scv[c] - scv[0]);
    const float fB = (float)B;
    atomicAdd(out + 0, l / (fB * (float)NNEG));
    atomicAdd(out + 1, red[0] / fB);
    atomicAdd(out + 2, red[1] / ((float)NN * fB));
    int indF[NC], indI[NC];
    argsort5_desc(sFv, indF);
    argsort5_desc(sIv, indI);
    const float lc = ssl2(userI, cand, indF) + ssl2(userF, cand, indI);
    atomicAdd(out + 3, lc / fB);
  }
}

extern "C" void kernel_launch(void* const* d_in, const int* in_sizes, int n_in,
                              void* d_out, int out_size, void* d_ws, size_t ws_size,
                              hipStream_t stream) {
  (void)n_in; (void)out_size; (void)ws_size;
  const float* emb      = (const float*)d_in[0];
  const float* W_PE     = (const float*)d_in[1];
  const float* W_pred   = (const float*)d_in[2];
  const float* W_gat    = (const float*)d_in[3];
  const float* a_src    = (const float*)d_in[4];
  const float* a_dst    = (const float*)d_in[5];
  const float* W_o      = (const float*)d_in[6];
  const float* W_assign = (const float*)d_in[7];
  const float* adj      = (const float*)d_in[8];
  const int*   history  = (const int*)d_in[9];
  const int*   timestp  = (const int*)d_in[10];
  const int*   pos_id   = (const int*)d_in[11];
  const int*   neg_id   = (const int*)d_in[12];
  const int*   stp      = (const int*)d_in[13];
  float* out = (float*)d_out;
  float* peW = (float*)d_ws;                 // 512*64 floats
  const int B = in_sizes[9] / NN;

  k_prep_pe<<<MAXLEN, 64, 0, stream>>>(W_PE, peW);
  k_init<<<1, 256, 0, stream>>>(W_pred, W_PE, out);
  k_main<<<B, 128, 0, stream>>>(emb, W_gat, a_src, a_dst, W_o, W_assign, adj,
                                W_pred, history, timestp, pos_id, neg_id, stp,
                                peW, out, B);
}